// GravNet_86483461472763
// MI455X (gfx1250) — compile-verified
//
#include <hip/hip_runtime.h>
#include <math.h>

typedef __attribute__((ext_vector_type(2))) float v2f;
typedef __attribute__((ext_vector_type(8))) float v8f;

#define NODES 8192
#define NPG   256
#define NG    32
#define KNN   20
#define KPMAX 260

// ---------------------------------------------------------------------------
// s = x@Ws + bs, h = x@Wh + bh; also copy x into padded concat buffer Xc
// Xc layout per row (stride Kp): [agg0, agg1, x[0..cin-1], 0-pad...]
// ---------------------------------------------------------------------------
__global__ void prep_sh_kernel(const float* __restrict__ x, int cin,
                               const float* __restrict__ Ws, const float* __restrict__ bs,
                               const float* __restrict__ Wh, const float* __restrict__ bh,
                               float* __restrict__ s, float* __restrict__ hp,
                               float* __restrict__ Xc, int Kp) {
    int n = blockIdx.x * blockDim.x + threadIdx.x;
    if (n >= NODES) return;
    const float* xr = x + (long)n * cin;
    float* xcr = Xc + (long)n * Kp;
    float s0 = 0.f, s1 = 0.f, s2 = 0.f, h0 = 0.f, h1 = 0.f;
    for (int c = 0; c < cin; ++c) {
        float xv = xr[c];
        s0 += xv * Ws[c * 3 + 0];
        s1 += xv * Ws[c * 3 + 1];
        s2 += xv * Ws[c * 3 + 2];
        h0 += xv * Wh[c * 2 + 0];
        h1 += xv * Wh[c * 2 + 1];
        xcr[2 + c] = xv;
    }
    for (int k = cin + 2; k < Kp; ++k) xcr[k] = 0.f;  // zero K-padding (every layer)
    s[n * 4 + 0] = s0 + bs[0];
    s[n * 4 + 1] = s1 + bs[1];
    s[n * 4 + 2] = s2 + bs[2];
    hp[n * 2 + 0] = h0 + bh[0];
    hp[n * 2 + 1] = h1 + bh[1];
}

// ---------------------------------------------------------------------------
// Per-graph kNN (top-20 smallest d2, includes self) + exp(-10*d2) aggregation.
// One block per graph; coords and h staged in LDS; top-20 held in registers
// (replace-max + fully unrolled bubble pass keeps arrays out of scratch).
// ---------------------------------------------------------------------------
__global__ __launch_bounds__(NPG) void knn_agg_kernel(const float* __restrict__ s,
                                                      const float* __restrict__ hp,
                                                      float* __restrict__ Xc, int Kp) {
    __shared__ float ss[NPG * 4];
    __shared__ float hh[NPG * 2];
    int g = blockIdx.x;
    int t = threadIdx.x;
    int n = g * NPG + t;
    ss[t * 4 + 0] = s[n * 4 + 0];
    ss[t * 4 + 1] = s[n * 4 + 1];
    ss[t * 4 + 2] = s[n * 4 + 2];
    hh[t * 2 + 0] = hp[n * 2 + 0];
    hh[t * 2 + 1] = hp[n * 2 + 1];
    __syncthreads();

    float a0 = ss[t * 4 + 0], a1 = ss[t * 4 + 1], a2 = ss[t * 4 + 2];
    float bd[KNN];
    int   bi[KNN];
#pragma unroll
    for (int k = 0; k < KNN; ++k) { bd[k] = 3.0e38f; bi[k] = 0; }

    for (int j = 0; j < NPG; ++j) {
        float d0 = a0 - ss[j * 4 + 0];
        float d1 = a1 - ss[j * 4 + 1];
        float d2 = a2 - ss[j * 4 + 2];
        float d = d0 * d0 + d1 * d1 + d2 * d2;
        if (d < bd[KNN - 1]) {
            bd[KNN - 1] = d;
            bi[KNN - 1] = j;
#pragma unroll
            for (int k = KNN - 1; k > 0; --k) {
                if (bd[k] < bd[k - 1]) {
                    float td = bd[k]; bd[k] = bd[k - 1]; bd[k - 1] = td;
                    int   ti = bi[k]; bi[k] = bi[k - 1]; bi[k - 1] = ti;
                }
            }
        }
    }

    float g0 = 0.f, g1 = 0.f;
#pragma unroll
    for (int k = 0; k < KNN; ++k) {
        float w = expf(-10.f * bd[k]);
        g0 += w * hh[bi[k] * 2 + 0];
        g1 += w * hh[bi[k] * 2 + 1];
    }
    float* xcr = Xc + (long)n * Kp;
    xcr[0] = g0;
    xcr[1] = g1;
}

// ---------------------------------------------------------------------------
// Zero-pad Wo[(cin+2) x cout] into Wp[Kp x cout]
// ---------------------------------------------------------------------------
__global__ void padw_kernel(const float* __restrict__ Wo, float* __restrict__ Wp,
                            int kin, int Kp, int cout) {
    int i = blockIdx.x * blockDim.x + threadIdx.x;
    if (i >= Kp * cout) return;
    int k = i / cout;
    int nn = i - k * cout;
    Wp[i] = (k < kin) ? Wo[k * cout + nn] : 0.f;
}

// ---------------------------------------------------------------------------
// out[M=8192, cout] = A[M, Kp] @ Wp[Kp, cout] + bo, optional LeakyReLU.
// Block = 8 waves x 32 lanes. Each wave owns a 16x16 (MxN) tile; block covers
// 128 rows x 16 cols. B tile (Kp x 16) staged in LDS. K-loop of f32 WMMAs.
// Fragment packing per ISA 7.12.2 (32-bit 16x4 A / 4x16 B layouts).
// ---------------------------------------------------------------------------
__global__ __launch_bounds__(256) void wmma_gemm_kernel(
        const float* __restrict__ A, const float* __restrict__ Wp,
        const float* __restrict__ bo, float* __restrict__ out,
        int Kp, int cout, int relu) {
    __shared__ float ldsB[KPMAX * 16];
    int lane = threadIdx.x;           // 0..31
    int wv = threadIdx.y;             // 0..7
    int tid = wv * 32 + lane;
    int nbase = blockIdx.y * 16;
    int mbase = blockIdx.x * 128 + wv * 16;

    for (int idx = tid; idx < Kp * 16; idx += 256) {
        int k = idx >> 4;
        int nn = idx & 15;
        ldsB[idx] = Wp[k * cout + nbase + nn];
    }
    __syncthreads();

    int half = lane >> 4;             // lanes 0-15 -> K pair {0,1}; 16-31 -> {2,3}
    int lr = lane & 15;               // A: row-in-tile; B/D: col-in-tile
    const float* arow = A + (long)(mbase + lr) * Kp + half * 2;

    v8f acc = {0.f, 0.f, 0.f, 0.f, 0.f, 0.f, 0.f, 0.f};
    for (int k0 = 0; k0 < Kp; k0 += 4) {
        v2f a;
        a.x = arow[k0];
        a.y = arow[k0 + 1];
        v2f b;
        b.x = ldsB[(k0 + half * 2) * 16 + lr];
        b.y = ldsB[(k0 + half * 2 + 1) * 16 + lr];
        acc = __builtin_amdgcn_wmma_f32_16x16x4_f32(
            /*neg_a=*/false, a, /*neg_b=*/false, b,
            /*c_mod=*/(short)0, acc, /*reuse_a=*/false, /*reuse_b=*/false);
    }

    float bias = bo[nbase + lr];
#pragma unroll
    for (int r = 0; r < 8; ++r) {
        int row = mbase + r + half * 8;   // D layout: VGPR r -> M=r (lanes<16), M=r+8
        float v = acc[r] + bias;
        if (relu) v = (v >= 0.f) ? v : 0.01f * v;
        out[(long)row * cout + nbase + lr] = v;
    }
}

// ---------------------------------------------------------------------------
// d_out[g*32 + c] = out3[(g*256) * 32 + c]  (first node of each graph)
// ---------------------------------------------------------------------------
__global__ void gather_kernel(const float* __restrict__ out3, float* __restrict__ dout) {
    int i = blockIdx.x * blockDim.x + threadIdx.x;
    if (i >= NG * 32) return;
    int g = i >> 5;
    int c = i & 31;
    dout[i] = out3[(long)(g * NPG) * 32 + c];
}

extern "C" void kernel_launch(void* const* d_in, const int* in_sizes, int n_in,
                              void* d_out, int out_size, void* d_ws, size_t ws_size,
                              hipStream_t stream) {
    const float* x   = (const float*)d_in[0];
    // d_in[1] = batch (int32) — graphs are contiguous blocks of 256, unused.
    const float* Ws1 = (const float*)d_in[2];  const float* bs1 = (const float*)d_in[3];
    const float* Wh1 = (const float*)d_in[4];  const float* bh1 = (const float*)d_in[5];
    const float* Wo1 = (const float*)d_in[6];  const float* bo1 = (const float*)d_in[7];
    const float* Ws2 = (const float*)d_in[8];  const float* bs2 = (const float*)d_in[9];
    const float* Wh2 = (const float*)d_in[10]; const float* bh2 = (const float*)d_in[11];
    const float* Wo2 = (const float*)d_in[12]; const float* bo2 = (const float*)d_in[13];
    const float* Ws3 = (const float*)d_in[14]; const float* bs3 = (const float*)d_in[15];
    const float* Wh3 = (const float*)d_in[16]; const float* bh3 = (const float*)d_in[17];
    const float* Wo3 = (const float*)d_in[18]; const float* bo3 = (const float*)d_in[19];

    float* ws   = (float*)d_ws;
    float* act1 = ws;                                // [8192*256]
    float* act2 = act1 + (size_t)NODES * 256;        // [8192*256]
    float* sbuf = act2 + (size_t)NODES * 256;        // [8192*4]
    float* hbuf = sbuf + (size_t)NODES * 4;          // [8192*2]
    float* Xc   = hbuf + (size_t)NODES * 2;          // [8192*260]
    float* Wp   = Xc + (size_t)NODES * KPMAX;        // [260*256]

    auto layer = [&](const float* xin, int cin,
                     const float* Ws, const float* bs,
                     const float* Wh, const float* bh,
                     const float* Wo, const float* bo,
                     int cout, int relu, float* outbuf) {
        int Kp = ((cin + 2) + 3) & ~3;               // 68 or 260
        prep_sh_kernel<<<NODES / 256, 256, 0, stream>>>(xin, cin, Ws, bs, Wh, bh,
                                                        sbuf, hbuf, Xc, Kp);
        knn_agg_kernel<<<NG, NPG, 0, stream>>>(sbuf, hbuf, Xc, Kp);
        int tot = Kp * cout;
        padw_kernel<<<(tot + 255) / 256, 256, 0, stream>>>(Wo, Wp, cin + 2, Kp, cout);
        dim3 grid(NODES / 128, cout / 16);
        dim3 blk(32, 8);
        wmma_gemm_kernel<<<grid, blk, 0, stream>>>(Xc, Wp, bo, outbuf, Kp, cout, relu);
    };

    layer(x,    64,  Ws1, bs1, Wh1, bh1, Wo1, bo1, 256, 1, act1);
    layer(act1, 256, Ws2, bs2, Wh2, bh2, Wo2, bo2, 256, 1, act2);
    layer(act2, 256, Ws3, bs3, Wh3, bh3, Wo3, bo3, 32,  0, act1);

    gather_kernel<<<(NG * 32 + 255) / 256, 256, 0, stream>>>(act1, (float*)d_out);
}